// Model_52948356825599
// MI455X (gfx1250) — compile-verified
//
#include <hip/hip_runtime.h>

typedef __attribute__((ext_vector_type(2))) float v2f;
typedef __attribute__((ext_vector_type(8))) float v8f;

#define TILE_K 32
#define HCOLS  128
#define WSP_PITCH 288   // 2*HCOLS + 32 pad: row stride mod 64 banks == 32 -> conflict-free b64 reads

// C[N,128] = (rowscale ? diag(1/max(cnt,1)) : I) * A[N,K] @ W[K,128]
//            (+ C if accumulate) (+ bias) (+ relu)
// Block: 256 threads = 8 waves, 128 rows of C. Wave: 16 rows x 128 cols via
// 8 accumulators of v_wmma_f32_16x16x4_f32.
__global__ __launch_bounds__(256)
void sage_gemm_kernel(const float* __restrict__ A, const float* __restrict__ W,
                      const float* __restrict__ bias, const float* __restrict__ cnt,
                      float* __restrict__ C, int N, int K,
                      int accumulate, int relu)
{
    __shared__ float As[8][16][TILE_K + 4];      // +4 pad: 16B aligned, conflict-free
    __shared__ float WsP[TILE_K / 2][WSP_PITCH]; // K-pair interleaved: [kp][col*2 + {0,1}]

    const int wave    = threadIdx.x >> 5;
    const int lane    = threadIdx.x & 31;
    const int rowBase = blockIdx.x * 128 + wave * 16;
    const int r16     = lane & 15;
    const int half    = lane >> 4;        // 0: K{0,1}, 1: K{2,3}
    const int halfK   = half << 1;

    v8f acc[8] = {};

    for (int k0 = 0; k0 < K; k0 += TILE_K) {
        // ---- W chunk [TILE_K x 128] -> LDS, interleaved by K-pair ----
        // 16 pairs x 128 cols; 512 float4-pair units; 256 threads x 2 iters.
        #pragma unroll
        for (int i = 0; i < 2; ++i) {
            int idx = threadIdx.x + i * 256;  // 0..511
            int kp  = idx >> 5;               // 0..15
            int c4  = idx & 31;               // float4 column
            const float* w0p = W + (size_t)(k0 + 2 * kp) * HCOLS + c4 * 4;
            float4 w0 = *(const float4*)(w0p);
            float4 w1 = *(const float4*)(w0p + HCOLS);
            float* o = &WsP[kp][c4 * 8];
            *(float4*)(o + 0) = make_float4(w0.x, w1.x, w0.y, w1.y);
            *(float4*)(o + 4) = make_float4(w0.z, w1.z, w0.w, w1.w);
        }
        // ---- per-wave A tile [16 x TILE_K] -> LDS ----
        #pragma unroll
        for (int i = 0; i < 4; ++i) {
            int idx  = lane + i * 32;         // 0..127; 8 float4 per row
            int r    = idx >> 3;
            int c4   = idx & 7;
            int grow = rowBase + r;
            float4 a = make_float4(0.f, 0.f, 0.f, 0.f);
            if (grow < N) {
                a = *(const float4*)(A + (size_t)grow * K + k0 + c4 * 4);
                if (cnt) {
                    float s = 1.0f / fmaxf(cnt[grow], 1.0f);
                    a.x *= s; a.y *= s; a.z *= s; a.w *= s;
                }
            }
            *(float4*)(&As[wave][r][c4 * 4]) = a;
        }
        // Prefetch next K-chunk while this one is consumed.
        if (k0 + TILE_K < K) {
            __builtin_prefetch(W + (size_t)(k0 + TILE_K + (threadIdx.x >> 3)) * HCOLS, 0, 1);
            int pr = rowBase + (lane >> 1);
            if (pr < N)
                __builtin_prefetch(A + (size_t)pr * K + k0 + TILE_K, 0, 1);
        }
        __syncthreads();

        #pragma unroll
        for (int kk = 0; kk < TILE_K; kk += 4) {
            // A frag: contiguous pair -> single b64 LDS read
            v2f a = *(const v2f*)(&As[wave][r16][kk + halfK]);
            const float* wrow = &WsP[(kk >> 1) + half][0];
            #pragma unroll
            for (int t = 0; t < 8; ++t) {
                // B frag: interleaved pair -> single b64 LDS read, no packing movs
                v2f b = *(const v2f*)(wrow + (t * 16 + r16) * 2);
                acc[t] = __builtin_amdgcn_wmma_f32_16x16x4_f32(
                    false, a, false, b, (short)0, acc[t], false, false);
            }
        }
        __syncthreads();
    }

    // C/D layout: VGPR v, lane L -> row = v + 8*(L/16), col = L%16 (+16*t)
    const int rOff = half * 8;
    #pragma unroll
    for (int t = 0; t < 8; ++t) {
        int   col = t * 16 + r16;
        float bv  = bias ? bias[col] : 0.f;
        #pragma unroll
        for (int v = 0; v < 8; ++v) {
            int grow = rowBase + rOff + v;
            if (grow < N) {
                float val = acc[t][v] + bv;
                size_t o = (size_t)grow * HCOLS + col;
                if (accumulate) val += C[o];
                if (relu) val = fmaxf(val, 0.f);
                C[o] = val;
            }
        }
    }
}

// One wave per edge: agg[dst] += h[src] (128 floats, float4/lane), cnt[dst] += 1.
__global__ __launch_bounds__(256)
void scatter_sum_kernel(const float* __restrict__ h, const int* __restrict__ src,
                        const int* __restrict__ dst, float* __restrict__ agg,
                        float* __restrict__ cnt, int E)
{
    int e    = blockIdx.x * 8 + (threadIdx.x >> 5);
    int lane = threadIdx.x & 31;
    if (e >= E) return;
    int s = src[e], d = dst[e];
    float4 v = *(const float4*)(h + (size_t)s * HCOLS + lane * 4);
    float* out = agg + (size_t)d * HCOLS + lane * 4;
    atomicAdd(out + 0, v.x);
    atomicAdd(out + 1, v.y);
    atomicAdd(out + 2, v.z);
    atomicAdd(out + 3, v.w);
    if (lane == 0) atomicAdd(cnt + d, 1.0f);
}

// One wave per edge: pred[e] = dot(h[src], h[dst]) over 128 cols.
__global__ __launch_bounds__(256)
void edge_dot_kernel(const float* __restrict__ h, const int* __restrict__ src,
                     const int* __restrict__ dst, float* __restrict__ pred, int E)
{
    int e    = blockIdx.x * 8 + (threadIdx.x >> 5);
    int lane = threadIdx.x & 31;
    if (e >= E) return;
    int s = src[e], d = dst[e];
    float4 a = *(const float4*)(h + (size_t)s * HCOLS + lane * 4);
    float4 b = *(const float4*)(h + (size_t)d * HCOLS + lane * 4);
    float p = a.x * b.x + a.y * b.y + a.z * b.z + a.w * b.w;
    #pragma unroll
    for (int off = 16; off > 0; off >>= 1)
        p += __shfl_down(p, off, 32);
    if (lane == 0) pred[e] = p;
}

extern "C" void kernel_launch(void* const* d_in, const int* in_sizes, int n_in,
                              void* d_out, int out_size, void* d_ws, size_t ws_size,
                              hipStream_t stream)
{
    const float* x      = (const float*)d_in[0];
    const int*   eidx   = (const int*)d_in[1];   // [2, E] int32
    const float* W_lin  = (const float*)d_in[2];
    const float* b_lin  = (const float*)d_in[3];
    const float* W_l1   = (const float*)d_in[4];
    const float* b_l1   = (const float*)d_in[5];
    const float* W_r1   = (const float*)d_in[6];
    const float* W_l2   = (const float*)d_in[7];
    const float* b_l2   = (const float*)d_in[8];
    const float* W_r2   = (const float*)d_in[9];

    const int D = 512, H = HCOLS;
    const int N = in_sizes[0] / D;
    const int E = in_sizes[1] / 2;
    const int* src = eidx;
    const int* dst = eidx + E;

    float* pred = (float*)d_out;          // [E]
    float* h2   = (float*)d_out + E;      // [N, H] (final h, part of output)

    // Workspace: h01 [N*H] (h0, overwritten in place by h1), agg [N*H], cnt [N]
    float* h01 = (float*)d_ws;
    float* agg = h01 + (size_t)N * H;
    float* cnt = agg + (size_t)N * H;

    dim3 blk(256);
    int gemmBlocks = (N + 127) / 128;
    int edgeBlocks = (E + 7) / 8;

    // h0 = x @ W_lin + b_lin
    sage_gemm_kernel<<<gemmBlocks, blk, 0, stream>>>(x, W_lin, b_lin, nullptr,
                                                     h01, N, D, 0, 0);
    // ---- SAGE layer 1 ----
    hipMemsetAsync(agg, 0, (size_t)N * H * sizeof(float), stream);
    hipMemsetAsync(cnt, 0, (size_t)N * sizeof(float), stream);
    scatter_sum_kernel<<<edgeBlocks, blk, 0, stream>>>(h01, src, dst, agg, cnt, E);
    // h1 = h0 @ W_r1   (in place: each block reads/writes only its own 128 rows)
    sage_gemm_kernel<<<gemmBlocks, blk, 0, stream>>>(h01, W_r1, nullptr, nullptr,
                                                     h01, N, H, 0, 0);
    // h1 = relu(h1 + mean @ W_l1 + b_l1)
    sage_gemm_kernel<<<gemmBlocks, blk, 0, stream>>>(agg, W_l1, b_l1, cnt,
                                                     h01, N, H, 1, 1);
    // ---- SAGE layer 2 ----
    hipMemsetAsync(agg, 0, (size_t)N * H * sizeof(float), stream);
    hipMemsetAsync(cnt, 0, (size_t)N * sizeof(float), stream);
    scatter_sum_kernel<<<edgeBlocks, blk, 0, stream>>>(h01, src, dst, agg, cnt, E);
    // h2 = h1 @ W_r2
    sage_gemm_kernel<<<gemmBlocks, blk, 0, stream>>>(h01, W_r2, nullptr, nullptr,
                                                     h2, N, H, 0, 0);
    // h2 = h2 + mean @ W_l2 + b_l2
    sage_gemm_kernel<<<gemmBlocks, blk, 0, stream>>>(agg, W_l2, b_l2, cnt,
                                                     h2, N, H, 1, 0);
    // ---- classifier: pred[e] = <h2[src], h2[dst]> ----
    edge_dot_kernel<<<edgeBlocks, blk, 0, stream>>>(h2, src, dst, pred, E);
}